// GPN_Encoder_73770358276678
// MI455X (gfx1250) — compile-verified
//
#include <hip/hip_runtime.h>

// GCN 2-layer forward for MI455X (gfx1250, wave32).
//
// Roofline: edge scatter (~2.6 GB random traffic, L2-resident since all node
// feature buffers < 192MB L2) dominates; GEMMs (13.1 + 3.3 GFLOP) run on the
// fp32 WMMA path (V_WMMA_F32_16X16X4_F32) for full reference precision.
// Scatter-add uses raw GLOBAL_ATOMIC_ADD_F32 (non-returning, device scope)
// via inline asm so we never fall back to a CAS retry loop.

typedef float v2f __attribute__((ext_vector_type(2)));
typedef float v8f __attribute__((ext_vector_type(8)));

#define NFEAT 512
#define NH1   256
#define NH2   128

// Single-instruction f32 atomic add at device scope (L2 RMW, no return).
__device__ __forceinline__ void atomic_fadd_dev(float* p, float v) {
#if defined(__gfx1250__)
  asm volatile("global_atomic_add_f32 %0, %1, off scope:SCOPE_DEV"
               :: "v"(p), "v"(v)
               : "memory");
#else
  atomicAdd(p, v);
#endif
}

// -------------------- degree / d^{-1/2} --------------------
__global__ void k_deg_init(float* __restrict__ deg, int n) {
  int i = blockIdx.x * blockDim.x + threadIdx.x;
  if (i < n) deg[i] = 1.0f;  // self-loop contributes 1 to every node's degree
}

__global__ void k_deg_count(const int* __restrict__ dst, int E, float* __restrict__ deg) {
  int e = blockIdx.x * blockDim.x + threadIdx.x;
  if (e < E) atomic_fadd_dev(&deg[dst[e]], 1.0f);
}

__global__ void k_dinv(float* __restrict__ deg, int n) {
  int i = blockIdx.x * blockDim.x + threadIdx.x;
  if (i < n) deg[i] = rsqrtf(deg[i]);  // deg >= 1 always (self-loops)
}

// -------------------- fp32 WMMA GEMM: C[MxN] = A[MxK] * B[KxN] --------------------
// Block: 256 threads = 8 waves arranged 4(M) x 2(N); block tile 64x64.
// Wave computes one 16(M) x 32(N) strip = 2 accumulators sharing the A fragment.
// K-chunks of 32 staged in LDS (A padded to stride 34 to spread banks).
#define APAD 34

__global__ __launch_bounds__(256)
void k_gemm_wmma(const float* __restrict__ A, const float* __restrict__ B,
                 float* __restrict__ C, int M, int K, int N) {
  __shared__ float As[64 * APAD];
  __shared__ float Bs[32 * 64];

  const int tid  = threadIdx.x;
  const int lane = tid & 31;
  const int wv   = tid >> 5;
  const int mw   = wv & 3;   // M sub-tile 0..3
  const int nw   = wv >> 2;  // N sub-tile 0..1
  const int hf   = lane >> 4;   // half-wave select (K pairing per ISA layout)
  const int idx  = lane & 15;

  const int m_base = blockIdx.y * 64;
  const int n_base = blockIdx.x * 64;

  v8f acc0 = {};
  v8f acc1 = {};

  for (int k0 = 0; k0 < K; k0 += 32) {
    // Stage A tile 64x32 (guard M edge; 50000 % 64 != 0)
#pragma unroll
    for (int i = 0; i < 8; ++i) {
      int eid = tid + i * 256;
      int r = eid >> 5, c = eid & 31;
      int gr = m_base + r;
      As[r * APAD + c] = (gr < M) ? A[(size_t)gr * K + k0 + c] : 0.0f;
    }
    // Stage B tile 32x64 (N, K always multiples of tile here)
#pragma unroll
    for (int i = 0; i < 8; ++i) {
      int eid = tid + i * 256;
      int r = eid >> 6, c = eid & 63;
      Bs[r * 64 + c] = B[(size_t)(k0 + r) * N + n_base + c];
    }
    __syncthreads();

#pragma unroll
    for (int kk = 0; kk < 32; kk += 4) {
      // A 16x4 fragment: v0 holds K=kk+2*hf, v1 holds K=kk+2*hf+1 (ISA 32-bit A layout)
      const float* ap = &As[(mw * 16 + idx) * APAD + kk + 2 * hf];
      v2f a; a.x = ap[0]; a.y = ap[1];
      // B 4x16 fragments (row striped across lanes): rows kk+2*hf, kk+2*hf+1
      const float* bp0 = &Bs[(kk + 2 * hf) * 64 + nw * 32 + idx];
      v2f b0; b0.x = bp0[0];  b0.y = bp0[64];
      v2f b1; b1.x = bp0[16]; b1.y = bp0[80];
      acc0 = __builtin_amdgcn_wmma_f32_16x16x4_f32(false, a, false, b0,
                                                   (short)0, acc0, false, false);
      acc1 = __builtin_amdgcn_wmma_f32_16x16x4_f32(false, a, false, b1,
                                                   (short)0, acc1, false, false);
    }
    __syncthreads();
  }

  // D layout: VGPR r -> M = r + 8*hf, N = idx
#pragma unroll
  for (int r = 0; r < 8; ++r) {
    int grow = m_base + mw * 16 + r + 8 * hf;
    if (grow < M) {
      C[(size_t)grow * N + n_base + nw * 32 + idx]      = acc0[r];
      C[(size_t)grow * N + n_base + nw * 32 + 16 + idx] = acc1[r];
    }
  }
}

// -------------------- edge gather * norm -> atomic scatter-add --------------------
// One wave per edge (incl. synthetic self-loop edges e >= E). Gather is float4
// coalesced; scatter uses single-instruction f32 global atomics (L2-resident dest).
template <int F>
__global__ __launch_bounds__(256)
void k_edge_scatter(const int* __restrict__ src, const int* __restrict__ dst,
                    int E, int n, const float* __restrict__ dinv,
                    const float* __restrict__ hin, float* __restrict__ hout) {
  int gw   = (blockIdx.x * blockDim.x + threadIdx.x) >> 5;
  int lane = threadIdx.x & 31;
  int total = E + n;
  if (gw >= total) return;

  int s, d;
  if (gw < E) { s = src[gw]; d = dst[gw]; }
  else        { s = d = gw - E; }
  float norm = dinv[s] * dinv[d];

  const float4* in4 = (const float4*)(hin + (size_t)s * F);
  float* outp = hout + (size_t)d * F;
#pragma unroll
  for (int g = 0; g < F / 128; ++g) {
    int c = lane + g * 32;          // float4 index, 0..F/4-1
    float4 v = in4[c];
    atomic_fadd_dev(outp + c * 4 + 0, v.x * norm);
    atomic_fadd_dev(outp + c * 4 + 1, v.y * norm);
    atomic_fadd_dev(outp + c * 4 + 2, v.z * norm);
    atomic_fadd_dev(outp + c * 4 + 3, v.w * norm);
  }
}

// -------------------- epilogues --------------------
__global__ void k_bias_relu(float* __restrict__ h, const float* __restrict__ b,
                            size_t total, int F) {
  size_t i = (size_t)blockIdx.x * blockDim.x + threadIdx.x;
  if (i < total) {
    float v = h[i] + b[i & (size_t)(F - 1)];
    h[i] = v > 0.0f ? v : 0.0f;
  }
}

__global__ void k_bias(float* __restrict__ h, const float* __restrict__ b,
                       size_t total, int F) {
  size_t i = (size_t)blockIdx.x * blockDim.x + threadIdx.x;
  if (i < total) h[i] += b[i & (size_t)(F - 1)];
}

// -------------------- driver --------------------
extern "C" void kernel_launch(void* const* d_in, const int* in_sizes, int n_in,
                              void* d_out, int out_size, void* d_ws, size_t ws_size,
                              hipStream_t stream) {
  const float* x  = (const float*)d_in[0];
  const int*   ei = (const int*)d_in[1];
  const float* W1 = (const float*)d_in[2];
  const float* b1 = (const float*)d_in[3];
  const float* W2 = (const float*)d_in[4];
  const float* b2 = (const float*)d_in[5];

  const int n = in_sizes[0] / NFEAT;
  const int E = in_sizes[1] / 2;
  const int* src = ei;
  const int* dst = ei + E;

  // Workspace layout (floats): dinv[n] | Hlin[n*256] | agg[n*256]
  // Hlin is reused for the layer-2 linear output (n*128 <= n*256).
  float* dinv = (float*)d_ws;
  float* Hlin = dinv + (size_t)((n + 255) & ~255);
  float* agg  = Hlin + (size_t)n * NH1;
  float* out  = (float*)d_out;

  // Degrees -> d^{-1/2}
  k_deg_init<<<(n + 255) / 256, 256, 0, stream>>>(dinv, n);
  k_deg_count<<<(E + 255) / 256, 256, 0, stream>>>(dst, E, dinv);
  k_dinv<<<(n + 255) / 256, 256, 0, stream>>>(dinv, n);

  const long long waves = (long long)E + n;
  const int scatter_blocks = (int)((waves * 32 + 255) / 256);

  // Layer 1: Hlin = x @ W1 ; agg = scatter(norm * gather(Hlin)) ; h1 = relu(agg + b1)
  k_gemm_wmma<<<dim3(NH1 / 64, (n + 63) / 64), 256, 0, stream>>>(x, W1, Hlin, n, NFEAT, NH1);
  hipMemsetAsync(agg, 0, (size_t)n * NH1 * sizeof(float), stream);
  k_edge_scatter<NH1><<<scatter_blocks, 256, 0, stream>>>(src, dst, E, n, dinv, Hlin, agg);
  k_bias_relu<<<(int)(((size_t)n * NH1 + 255) / 256), 256, 0, stream>>>(
      agg, b1, (size_t)n * NH1, NH1);

  // Layer 2: Hlin = h1 @ W2 ; out = scatter(...) + b2
  k_gemm_wmma<<<dim3(NH2 / 64, (n + 63) / 64), 256, 0, stream>>>(agg, W2, Hlin, n, NH1, NH2);
  hipMemsetAsync(out, 0, (size_t)out_size * sizeof(float), stream);
  k_edge_scatter<NH2><<<scatter_blocks, 256, 0, stream>>>(src, dst, E, n, dinv, Hlin, out);
  k_bias<<<(int)(((size_t)out_size + 255) / 256), 256, 0, stream>>>(
      out, b2, (size_t)out_size, NH2);
}